// ZZ_X_Z_circuit_16484084483309
// MI455X (gfx1250) — compile-verified
//
#include <hip/hip_runtime.h>
#include <hip/hip_bf16.h>
#include <stdint.h>

// NQ=22 qubits; all gates act on the low 11 index bits -> 2048 independent
// contiguous blocks of 2048 complex amps under one shared 2048x2048 unitary.
// Per layer: 11 fused RZ*RX single-qubit gates == 3 grouped complex 16x16
// Kronecker matrices (bits 10-7, 6-3, I2(x)G on 3-0) -> 3 GEMM passes of
// G(16x16) x X(16x128) done with chained V_WMMA_F32_16X16X4_F32; the ZZ chain
// is a diagonal fused into pass-C stores via a per-WG phase table.

#define QSIZE    11
#define QBLOCK   2048
#define QTHREADS 256
#define QNWG     256
#define QBPW     (2048 / QNWG)     // blocks per workgroup

typedef float v8f __attribute__((ext_vector_type(8)));
typedef float v2f __attribute__((ext_vector_type(2)));

__device__ __forceinline__ v8f wmma4(v2f a, v2f b, v8f c) {
    return __builtin_amdgcn_wmma_f32_16x16x4_f32(false, a, false, b, (short)0, c, false, false);
}

__device__ __forceinline__ void load_afrag(const float* af, v2f ar[4], v2f ai[4]) {
    const float4 f0 = *(const float4*)(af + 0);
    const float4 f1 = *(const float4*)(af + 4);
    const float4 f2 = *(const float4*)(af + 8);
    const float4 f3 = *(const float4*)(af + 12);
    ar[0].x = f0.x; ar[0].y = f0.y;  ar[1].x = f0.z; ar[1].y = f0.w;
    ar[2].x = f1.x; ar[2].y = f1.y;  ar[3].x = f1.z; ar[3].y = f1.w;
    ai[0].x = f2.x; ai[0].y = f2.y;  ai[1].x = f2.z; ai[1].y = f2.w;
    ai[2].x = f3.x; ai[2].y = f3.y;  ai[3].x = f3.z; ai[3].y = f3.w;
}

__global__ __launch_bounds__(QTHREADS) void ZZ_X_Z_circuit_kernel(
    const float* __restrict__ theta,   // 64 params
    const float* __restrict__ s_in,    // 2^22 real fp32
    float* __restrict__ out)           // 2^23 fp32 (complex64 interleaved)
{
    __shared__ float  sre[QBLOCK];
    __shared__ float  sim[QBLOCK];
    __shared__ float  M[2][QSIZE][8];                 // fused RZ*RX 2x2 (r,i pairs)
    __shared__ __align__(16) float2 PH[2][QBLOCK];    // ZZ phase tables
    __shared__ __align__(16) float  AF[6][32][20];    // per-lane A-fragments (padded)

    const int t    = threadIdx.x;
    const int lane = t & 31;
    const int w    = t >> 5;          // wave id = column tile
    const int n    = lane & 15;       // A row / B,D column
    const int h    = lane >> 4;       // lane half -> K slot 2h+j

    // ---- setup 1: fused RZ*RX 2x2 matrices (threads 0..21) ----
    if (t < 2 * QSIZE) {
        const int layer = t / QSIZE, i = t % QSIZE;
        const float thx = theta[layer * 32 + 2 * i];
        const float thz = theta[layer * 32 + 2 * i + 1];
        float c, sn, cz, sz;
        __sincosf(0.5f * thx, &sn, &c);
        __sincosf(0.5f * thz, &sz, &cz);
        float* m = M[layer][i];
        m[0] =  cz * c;  m[1] = -sz * c;   // m00 = e^{-iz} c
        m[2] = -sn * sz; m[3] = -sn * cz;  // m01 = e^{-iz} (-i s)
        m[4] =  sn * sz; m[5] = -sn * cz;  // m10 = e^{+iz} (-i s)
        m[6] =  cz * c;  m[7] =  sz * c;   // m11 = e^{+iz} c
    }
    // ---- setup 1b: ZZ phase tables (all threads) ----
    for (int layer = 0; layer < 2; ++layer) {
        float zh[QSIZE - 1];
        #pragma unroll
        for (int i = 0; i < QSIZE - 1; ++i)
            zh[i] = 0.5f * theta[layer * 32 + 2 * QSIZE + i];
        #pragma unroll
        for (int k = 0; k < QBLOCK / QTHREADS; ++k) {
            const int l = t + k * QTHREADS;
            const unsigned x = (unsigned)(l ^ (l >> 1));
            float ang = 0.f;
            #pragma unroll
            for (int i = 0; i < QSIZE - 1; ++i)
                ang += (((x >> (9 - i)) & 1u) ? -zh[i] : zh[i]);
            float sA, cA;
            __sincosf(ang, &sA, &cA);
            PH[layer][l] = make_float2(cA, sA);
        }
    }
    __syncthreads();

    // ---- setup 2: A-fragments for the 6 passes (wave 0 only) ----
    // Group g row/col index bits map to wires: gA wires 0-3 (bit3->wire0),
    // gB wires 4-7, gC = I2 (x) (wires 8-10). Entry = product of 2x2 entries.
    if (t < 32) {
        for (int pi = 0; pi < 6; ++pi) {
            const int L = pi / 3, g = pi % 3;
            #pragma unroll
            for (int kc = 0; kc < 4; ++kc) {
                #pragma unroll
                for (int j = 0; j < 2; ++j) {
                    const int kk = 4 * kc + 2 * h + j;
                    float pr = 1.f, pim = 0.f;
                    if (g < 2) {
                        #pragma unroll
                        for (int q = 0; q < 4; ++q) {
                            const int rb = (n  >> (3 - q)) & 1;
                            const int cb = (kk >> (3 - q)) & 1;
                            const float* e = &M[L][g * 4 + q][(rb * 2 + cb) * 2];
                            const float nr = pr * e[0] - pim * e[1];
                            const float ni = pr * e[1] + pim * e[0];
                            pr = nr; pim = ni;
                        }
                    } else if (((n ^ kk) >> 3) & 1) {
                        pr = 0.f; pim = 0.f;          // I2 block off-diagonal
                    } else {
                        #pragma unroll
                        for (int q = 0; q < 3; ++q) {
                            const int rb = (n  >> (2 - q)) & 1;
                            const int cb = (kk >> (2 - q)) & 1;
                            const float* e = &M[L][8 + q][(rb * 2 + cb) * 2];
                            const float nr = pr * e[0] - pim * e[1];
                            const float ni = pr * e[1] + pim * e[0];
                            pr = nr; pim = ni;
                        }
                    }
                    AF[pi][lane][2 * kc + j]     = pr;
                    AF[pi][lane][8 + 2 * kc + j] = pim;
                }
            }
        }
    }
    __syncthreads();

    const int cc   = w * 16 + n;                       // global column of this lane
    const int colA = cc;                               // rstride 128
    const int colB = (cc >> 3) * 128 + (cc & 7);       // rstride 8
    const int colC = (cc >> 3) * 128 + (cc & 7) * 16;  // rstride 1

    #pragma unroll 1
    for (int blk = 0; blk < QBPW; ++blk) {
        const int hblk = blockIdx.x * QBPW + blk;
        const float* gin = s_in + (size_t)hblk * QBLOCK;
        float* gout      = out  + (size_t)hblk * (2 * QBLOCK);

        // ---- L0 pass A: real input from global (imag=0 -> 8 WMMAs) ----
        {
            v2f ar[4], ai[4]; load_afrag(&AF[0][lane][0], ar, ai);
            v2f br[4];
            #pragma unroll
            for (int kc = 0; kc < 4; ++kc) {
                const int k0 = 4 * kc + 2 * h;
                br[kc].x = gin[colA + k0 * 128];
                br[kc].y = gin[colA + (k0 + 1) * 128];
            }
            v8f cr = {}, ci = {};
            #pragma unroll
            for (int kc = 0; kc < 4; ++kc) {
                cr = wmma4(ar[kc], br[kc], cr);
                ci = wmma4(ai[kc], br[kc], ci);
            }
            #pragma unroll
            for (int r = 0; r < 8; ++r) {
                const int l = colA + (r + 8 * h) * 128;
                sre[l] = cr[r]; sim[l] = ci[r];
            }
        }
        __syncthreads();

        // ---- generic complex LDS passes (L0B, then L1A, L1B) + phase passes ----
        #pragma unroll 1
        for (int layer = 0; layer < 2; ++layer) {
            if (layer == 1) {  // L1 pass A (complex, from LDS, rstride 128)
                v2f ar[4], ai[4]; load_afrag(&AF[3][lane][0], ar, ai);
                v2f br[4], bi[4];
                #pragma unroll
                for (int kc = 0; kc < 4; ++kc) {
                    const int e0 = colA + (4 * kc + 2 * h) * 128;
                    br[kc].x = sre[e0]; br[kc].y = sre[e0 + 128];
                    bi[kc].x = sim[e0]; bi[kc].y = sim[e0 + 128];
                }
                v8f crp = {}, crm = {}, ca = {}, cb = {};
                #pragma unroll
                for (int kc = 0; kc < 4; ++kc) {
                    crp = wmma4(ar[kc], br[kc], crp);
                    crm = wmma4(ai[kc], bi[kc], crm);
                    ca  = wmma4(ar[kc], bi[kc], ca);
                    cb  = wmma4(ai[kc], br[kc], cb);
                }
                const v8f cr = crp - crm, ci = ca + cb;
                #pragma unroll
                for (int r = 0; r < 8; ++r) {
                    const int l = colA + (r + 8 * h) * 128;
                    sre[l] = cr[r]; sim[l] = ci[r];
                }
                __syncthreads();
            }
            // ---- pass B (complex, rstride 8) ----
            {
                v2f ar[4], ai[4]; load_afrag(&AF[layer * 3 + 1][lane][0], ar, ai);
                v2f br[4], bi[4];
                #pragma unroll
                for (int kc = 0; kc < 4; ++kc) {
                    const int e0 = colB + (4 * kc + 2 * h) * 8;
                    br[kc].x = sre[e0]; br[kc].y = sre[e0 + 8];
                    bi[kc].x = sim[e0]; bi[kc].y = sim[e0 + 8];
                }
                v8f crp = {}, crm = {}, ca = {}, cb = {};
                #pragma unroll
                for (int kc = 0; kc < 4; ++kc) {
                    crp = wmma4(ar[kc], br[kc], crp);
                    crm = wmma4(ai[kc], bi[kc], crm);
                    ca  = wmma4(ar[kc], bi[kc], ca);
                    cb  = wmma4(ai[kc], br[kc], cb);
                }
                const v8f cr = crp - crm, ci = ca + cb;
                #pragma unroll
                for (int r = 0; r < 8; ++r) {
                    const int l = colB + (r + 8 * h) * 8;
                    sre[l] = cr[r]; sim[l] = ci[r];
                }
                __syncthreads();
            }
            // ---- pass C (complex, rstride 1) + fused ZZ phase ----
            {
                v2f ar[4], ai[4]; load_afrag(&AF[layer * 3 + 2][lane][0], ar, ai);
                v2f br[4], bi[4];
                #pragma unroll
                for (int kc = 0; kc < 4; ++kc) {
                    const int e0 = colC + 4 * kc + 2 * h;
                    br[kc].x = sre[e0]; br[kc].y = sre[e0 + 1];
                    bi[kc].x = sim[e0]; bi[kc].y = sim[e0 + 1];
                }
                v8f crp = {}, crm = {}, ca = {}, cb = {};
                #pragma unroll
                for (int kc = 0; kc < 4; ++kc) {
                    crp = wmma4(ar[kc], br[kc], crp);
                    crm = wmma4(ai[kc], bi[kc], crm);
                    ca  = wmma4(ar[kc], bi[kc], ca);
                    cb  = wmma4(ai[kc], br[kc], cb);
                }
                const v8f cr = crp - crm, ci = ca + cb;
                const int base = colC + 8 * h;
                const float4* php = (const float4*)&PH[layer][base];
                if (layer == 0) {          // rotate, keep in LDS
                    #pragma unroll
                    for (int p = 0; p < 4; ++p) {
                        const float4 q = php[p];
                        sre[base + 2 * p]     = cr[2 * p] * q.x - ci[2 * p] * q.y;
                        sim[base + 2 * p]     = cr[2 * p] * q.y + ci[2 * p] * q.x;
                        sre[base + 2 * p + 1] = cr[2 * p + 1] * q.z - ci[2 * p + 1] * q.w;
                        sim[base + 2 * p + 1] = cr[2 * p + 1] * q.w + ci[2 * p + 1] * q.z;
                    }
                } else {                    // rotate, store straight to global
                    float4* g4 = (float4*)(gout + 2 * base);
                    #pragma unroll
                    for (int p = 0; p < 4; ++p) {
                        const float4 q = php[p];
                        float4 o;
                        o.x = cr[2 * p] * q.x - ci[2 * p] * q.y;
                        o.y = cr[2 * p] * q.y + ci[2 * p] * q.x;
                        o.z = cr[2 * p + 1] * q.z - ci[2 * p + 1] * q.w;
                        o.w = cr[2 * p + 1] * q.w + ci[2 * p + 1] * q.z;
                        g4[p] = o;
                    }
                }
                __syncthreads();   // also protects LDS reuse by next block
            }
        }
    }
}

extern "C" void kernel_launch(void* const* d_in, const int* in_sizes, int n_in,
                              void* d_out, int out_size, void* d_ws, size_t ws_size,
                              hipStream_t stream) {
    const float* theta = (const float*)d_in[0];
    const float* s     = (const float*)d_in[1];
    float* out         = (float*)d_out;
    ZZ_X_Z_circuit_kernel<<<QNWG, QTHREADS, 0, stream>>>(theta, s, out);
}